// NodeTokenAdapter_84799834292888
// MI455X (gfx1250) — compile-verified
//
#include <hip/hip_runtime.h>
#include <hip/hip_bf16.h>
#include <math.h>

typedef __attribute__((ext_vector_type(2))) float v2f;
typedef __attribute__((ext_vector_type(8))) float v8f;

#define WPB 8            // waves per block (wave32)
#define TPB (WPB * 32)   // 256 threads

// ---------------- small helper kernels ----------------

__global__ void zero_counts_kernel(const int* bsz, int* ws) {
    int B = *bsz;
    int i = blockIdx.x * blockDim.x + threadIdx.x;
    if (i < B) ws[i] = 0;
}

__global__ void hist_kernel(const int* __restrict__ bidx, int N, int* __restrict__ ws) {
    int i = blockIdx.x * blockDim.x + threadIdx.x;
    if (i < N) atomicAdd(&ws[bidx[i]], 1);
}

// Single-block Hillis-Steele inclusive scan -> exclusive "starts" at ws[B..2B)
__global__ void scan_kernel(const int* bsz, int* ws) {
    __shared__ int s[1024];
    int B = *bsz;                 // assume B <= 1024 (setup uses 1024)
    int t = threadIdx.x;
    int v = (t < B) ? ws[t] : 0;
    s[t] = v;
    __syncthreads();
    for (int off = 1; off < 1024; off <<= 1) {
        int x = (t >= off) ? s[t - off] : 0;
        __syncthreads();
        s[t] += x;
        __syncthreads();
    }
    if (t < B) ws[B + t] = s[t] - v;   // exclusive prefix = group start
}

// Zero padded token rows, write int32 enc_mask. grid = rows (= B*T), block = 128.
__global__ void pad_mask_kernel(const int* bsz, const int* __restrict__ ws,
                                float* __restrict__ out, long long rows) {
    int B = *bsz;
    int T = (int)(rows / B);
    long long row = blockIdx.x;
    int b = (int)(row / T);
    int t = (int)(row - (long long)b * T);
    int c = ws[b];                       // counts
    int valid = (t < c) ? 1 : 0;
    if (!valid) out[row * 128 + threadIdx.x] = 0.0f;
    if (threadIdx.x == 0) {
        int* maskp = (int*)out + rows * 128LL;   // int32 mask region
        maskp[row] = valid;
    }
}

// ---------------- fused GEMM + GELU + LayerNorm + scatter ----------------

__device__ inline long long shfl_ll(long long v, int src) {
    int lo = __shfl((int)(v & 0xffffffffLL), src, 32);
    int hi = __shfl((int)(v >> 32), src, 32);
    return ((long long)hi << 32) | (unsigned int)lo;
}

__global__ __launch_bounds__(TPB)
void fused_mlp_ln_scatter(const float* __restrict__ x,      // [N,128]
                          const int*   __restrict__ bidx,   // [N] sorted
                          const int*   bsz,
                          const float* __restrict__ W,      // [128,128] row-major (k,n)
                          const float* __restrict__ bias,   // [128]
                          const float* __restrict__ gamma,  // [128]
                          const float* __restrict__ beta,   // [128]
                          const int*   __restrict__ ws,     // counts[B], starts[B]
                          float*       __restrict__ out,    // tokens then mask
                          long long rows, int N) {
    int B = *bsz;
    int T = (int)(rows / B);
    const int* starts = ws + B;

    // Swizzled W fragments in LDS: entry (nt, k, lane) = { W[kr][col], W[kr+1][col] }
    // with kr = 4k + 2*(lane>>4), col = nt*16 + (lane&15)  -> one ds_load_b64 per WMMA.
    __shared__ v2f lws[8 * 32 * 32];   // 64 KB

    for (int e = threadIdx.x; e < 8 * 32 * 32; e += blockDim.x) {
        int nt = e >> 10;
        int k  = (e >> 5) & 31;
        int ln = e & 31;
        int hh = ln >> 4, mm = ln & 15;
        int kr  = 4 * k + 2 * hh;
        int col = nt * 16 + mm;
        v2f w;
        w.x = W[kr * 128 + col];
        w.y = W[(kr + 1) * 128 + col];
        lws[e] = w;
    }
    __syncthreads();

    int lane = threadIdx.x & 31;
    int wave = threadIdx.x >> 5;
    int m  = lane & 15;
    int hi = lane >> 4;
    int numTiles = (N + 15) >> 4;
    int waveId = blockIdx.x * WPB + wave;
    int nWaves = gridDim.x * WPB;

    for (int tile = waveId; tile < numTiles; tile += nWaves) {
        int node0 = tile << 4;
        int nrow = node0 + m; if (nrow > N - 1) nrow = N - 1;
        const float* xrow = x + (size_t)nrow * 128 + 2 * hi;

        v8f acc[8] = {};   // 8 column tiles of the 16x128 output

        for (int k = 0; k < 32; ++k) {
            v2f a = *(const v2f*)(xrow + 4 * k);   // A fragment: rows striped per ISA layout
            #pragma unroll
            for (int nt = 0; nt < 8; ++nt) {
                v2f bw = lws[(nt << 10) + (k << 5) + lane];
                acc[nt] = __builtin_amdgcn_wmma_f32_16x16x4_f32(
                    false, a, false, bw, (short)0, acc[nt], false, false);
            }
        }

        // bias + exact-erf GELU; per-row sums for LayerNorm.
        // Lane's element r of acc[nt] is (row = r + 8*hi, col = nt*16 + m).
        float g[8][8];
        float rsum[8], rsq[8];
        #pragma unroll
        for (int r = 0; r < 8; ++r) { rsum[r] = 0.f; rsq[r] = 0.f; }
        #pragma unroll
        for (int nt = 0; nt < 8; ++nt) {
            int col = nt * 16 + m;
            float bb = bias[col];
            #pragma unroll
            for (int r = 0; r < 8; ++r) {
                float v  = acc[nt][r] + bb;
                float gg = 0.5f * v * (1.0f + erff(v * 0.70710678118654752f));
                g[nt][r] = gg;
                rsum[r] += gg;
                rsq[r]  += gg * gg;
            }
        }
        // Row reduction within each 16-lane half (rows 0-7 in lanes 0-15, 8-15 in 16-31)
        #pragma unroll
        for (int r = 0; r < 8; ++r) {
            #pragma unroll
            for (int off = 1; off < 16; off <<= 1) {
                rsum[r] += __shfl_xor(rsum[r], off, 32);
                rsq[r]  += __shfl_xor(rsq[r],  off, 32);
            }
        }
        float mean[8], rstd[8];
        #pragma unroll
        for (int r = 0; r < 8; ++r) {
            float mu  = rsum[r] * (1.0f / 128.0f);
            float var = rsq[r]  * (1.0f / 128.0f) - mu * mu;
            mean[r] = mu;
            rstd[r] = rsqrtf(var + 1e-5f);
        }

        // Scatter addressing: lane m holds the row-base for node node0+m.
        long long rowbase = -1;
        {
            int iN = node0 + m;
            if (iN < N) {
                int bI  = bidx[iN];
                int pos = iN - starts[bI];
                rowbase = ((long long)bI * T + pos) * 128LL;
            }
        }
        long long rb[8];
        #pragma unroll
        for (int r = 0; r < 8; ++r) rb[r] = shfl_ll(rowbase, r + 8 * hi);

        #pragma unroll
        for (int nt = 0; nt < 8; ++nt) {
            int col = nt * 16 + m;
            float ga = gamma[col], be = beta[col];
            #pragma unroll
            for (int r = 0; r < 8; ++r) {
                if (rb[r] >= 0)
                    out[rb[r] + col] = (g[nt][r] - mean[r]) * rstd[r] * ga + be;
            }
        }
    }
}

// ---------------- launch ----------------

extern "C" void kernel_launch(void* const* d_in, const int* in_sizes, int n_in,
                              void* d_out, int out_size, void* d_ws, size_t ws_size,
                              hipStream_t stream) {
    const float* node_emb = (const float*)d_in[0];
    const int*   bidx     = (const int*)d_in[1];
    const int*   bsz      = (const int*)d_in[2];   // device scalar B
    const float* W        = (const float*)d_in[3];
    const float* bias     = (const float*)d_in[4];
    const float* gamma    = (const float*)d_in[5];
    const float* beta     = (const float*)d_in[6];
    float* out = (float*)d_out;
    int*   ws  = (int*)d_ws;

    int N = in_sizes[1];
    long long rows = (long long)out_size / 129;    // B*T (tokens rows)

    zero_counts_kernel<<<16, 256, 0, stream>>>(bsz, ws);
    hist_kernel<<<(N + 255) / 256, 256, 0, stream>>>(bidx, N, ws);
    scan_kernel<<<1, 1024, 0, stream>>>(bsz, ws);

    int numTiles = (N + 15) / 16;
    int blocks = (numTiles + WPB - 1) / WPB;
    if (blocks > 1280) blocks = 1280;
    fused_mlp_ln_scatter<<<blocks, TPB, 0, stream>>>(node_emb, bidx, bsz, W, bias,
                                                     gamma, beta, ws, out, rows, N);

    pad_mask_kernel<<<(unsigned)rows, 128, 0, stream>>>(bsz, ws, out, rows);
}